// HBNS_89275190214711
// MI455X (gfx1250) — compile-verified
//
#include <hip/hip_runtime.h>

typedef __attribute__((ext_vector_type(2))) float v2f;
typedef __attribute__((ext_vector_type(8))) float v8f;

#define K_IN      128
#define N_OUT     64
#define NEG_SLOPE 0.2f

// ---------------------------------------------------------------------------
// Y[n_rows x 64] = X[n_rows x 128] @ W[128 x 64], full fp32 via
// V_WMMA_F32_16X16X4_F32. One wave per 16-row tile; each wave accumulates the
// full 16x64 output tile (4 N-tiles) across K=128 (32 WMMA issues x 4 tiles).
// ---------------------------------------------------------------------------
__global__ __launch_bounds__(256) void gemm_wmma_f32(
    const float* __restrict__ X, const float* __restrict__ W,
    float* __restrict__ Y, int n_rows) {
  const int lane = threadIdx.x & 31;
  const int wv   = threadIdx.x >> 5;
  const int tile = blockIdx.x * 8 + wv;
  const int row0 = tile * 16;
  if (row0 >= n_rows) return;

  const int m  = lane & 15;   // row within tile (A) / column within N-tile (B,C)
  const int kh = lane >> 4;   // half-wave: K offset 0 or 2

  const float* xrow = X + (size_t)(row0 + m) * K_IN;

  v8f acc0 = {}, acc1 = {}, acc2 = {}, acc3 = {};

  for (int k = 0; k < K_IN; k += 4) {
    const int kk = k + 2 * kh;
    // A fragment: lane holds (row m, K = kk, kk+1) -> contiguous float2
    const float2 av = *(const float2*)(xrow + kk);
    v2f a; a.x = av.x; a.y = av.y;

    // B fragments: VGPR0 = K-row kk, VGPR1 = K-row kk+1, col = ntile*16 + m
    const float* wp = W + (size_t)kk * N_OUT + m;
    v2f b0, b1, b2, b3;
    b0.x = wp[0];  b0.y = wp[N_OUT + 0];
    b1.x = wp[16]; b1.y = wp[N_OUT + 16];
    b2.x = wp[32]; b2.y = wp[N_OUT + 32];
    b3.x = wp[48]; b3.y = wp[N_OUT + 48];

    acc0 = __builtin_amdgcn_wmma_f32_16x16x4_f32(false, a, false, b0, (short)0, acc0, false, false);
    acc1 = __builtin_amdgcn_wmma_f32_16x16x4_f32(false, a, false, b1, (short)0, acc1, false, false);
    acc2 = __builtin_amdgcn_wmma_f32_16x16x4_f32(false, a, false, b2, (short)0, acc2, false, false);
    acc3 = __builtin_amdgcn_wmma_f32_16x16x4_f32(false, a, false, b3, (short)0, acc3, false, false);
  }

  // C/D layout: VGPR v -> row (v + 8*kh), col = m (+ 16*ntile)
  float* yb = Y + (size_t)(row0 + 8 * kh) * N_OUT + m;
#pragma unroll
  for (int v = 0; v < 8; ++v) {
    yb[(size_t)v * N_OUT + 0]  = acc0[v];
    yb[(size_t)v * N_OUT + 16] = acc1[v];
    yb[(size_t)v * N_OUT + 32] = acc2[v];
    yb[(size_t)v * N_OUT + 48] = acc3[v];
  }
}

// out[r] = dot(M[r, 0:64], w[0:64])
__global__ void rowdot64(const float* __restrict__ M, const float* __restrict__ w,
                         float* __restrict__ out, int n_rows) {
  int r = blockIdx.x * blockDim.x + threadIdx.x;
  if (r >= n_rows) return;
  const float4* mp = (const float4*)(M + (size_t)r * N_OUT);
  const float4* wp = (const float4*)w;
  float s = 0.f;
#pragma unroll
  for (int i = 0; i < 16; ++i) {
    float4 a = mp[i], b = wp[i];
    s += a.x * b.x + a.y * b.y + a.z * b.z + a.w * b.w;
  }
  out[r] = s;
}

// Grid-stride zero of two float regions.
__global__ void zero_two(float* __restrict__ a, long na,
                         float* __restrict__ b, long nb) {
  long i = blockIdx.x * (long)blockDim.x + threadIdx.x;
  long stride = (long)gridDim.x * blockDim.x;
  for (long j = i; j < na; j += stride) a[j] = 0.f;
  for (long j = i; j < nb; j += stride) b[j] = 0.f;
}

// Per-edge leaky-relu score (f == e since a_s==b_s, a_t==b_t) + both
// segment-sum denominators via fp32 atomics (resolve in L2).
__global__ void edge_score(const int* __restrict__ rows, const int* __restrict__ cols,
                           const float* __restrict__ es, const float* __restrict__ et,
                           float* __restrict__ e_buf,
                           float* __restrict__ e_den, float* __restrict__ f_den,
                           int n_edges) {
  int i = blockIdx.x * blockDim.x + threadIdx.x;
  if (i >= n_edges) return;
  const int r = rows[i], c = cols[i];
  float v = es[c] + et[r];
  v = (v > 0.f) ? v : v * NEG_SLOPE;
  e_buf[i] = v;
  atomicAdd(&e_den[r], v);
  atomicAdd(&f_den[c], v);
}

// One wave32 per edge: lanes cover 64 columns as float2 pairs.
// msg_tgt[r] += (e/e_den[r])*nbhd * s_msg[c];  msg_src[c] += (e/f_den[c])*nbhd * t_msg[r]
__global__ __launch_bounds__(256) void edge_aggregate(
    const int* __restrict__ rows, const int* __restrict__ cols,
    const float* __restrict__ nbhd, const float* __restrict__ e_buf,
    const float* __restrict__ e_den, const float* __restrict__ f_den,
    const float* __restrict__ s_msg, const float* __restrict__ t_msg,
    float* __restrict__ msg_src, float* __restrict__ msg_tgt, int n_edges) {
  const int lane = threadIdx.x & 31;
  const int wv   = threadIdx.x >> 5;
  const int edge = blockIdx.x * 8 + wv;
  if (edge >= n_edges) return;

  const int r = rows[edge], c = cols[edge];
  const float ev = e_buf[edge];
  const float nb = nbhd[edge];
  const float w_st = ev / e_den[r] * nb;
  const float w_ts = ev / f_den[c] * nb;

  const float2 s = ((const float2*)(s_msg + (size_t)c * N_OUT))[lane];
  const float2 t = ((const float2*)(t_msg + (size_t)r * N_OUT))[lane];

  float* pt = msg_tgt + (size_t)r * N_OUT + 2 * lane;
  float* ps = msg_src + (size_t)c * N_OUT + 2 * lane;
  atomicAdd(pt + 0, w_st * s.x);
  atomicAdd(pt + 1, w_st * s.y);
  atomicAdd(ps + 0, w_ts * t.x);
  atomicAdd(ps + 1, w_ts * t.y);
}

extern "C" void kernel_launch(void* const* d_in, const int* in_sizes, int n_in,
                              void* d_out, int out_size, void* d_ws, size_t ws_size,
                              hipStream_t stream) {
  const float* x_source = (const float*)d_in[0];
  const float* x_target = (const float*)d_in[1];
  const float* w_s      = (const float*)d_in[2];
  const float* w_t      = (const float*)d_in[3];
  const float* att      = (const float*)d_in[4];
  const float* nbhd     = (const float*)d_in[5];
  const int*   rows     = (const int*)d_in[6];
  const int*   cols     = (const int*)d_in[7];

  const int n_src   = in_sizes[0] / K_IN;   // 100000
  const int n_tgt   = in_sizes[1] / K_IN;   // 50000
  const int n_edges = in_sizes[5];          // 1000000

  // Outputs: (message_on_source [n_src x 64], message_on_target [n_tgt x 64])
  float* msg_src = (float*)d_out;
  float* msg_tgt = msg_src + (size_t)n_src * N_OUT;

  // Workspace layout (floats): s_msg | t_msg | es | et | e_den | f_den | e_buf
  float* ws    = (float*)d_ws;
  float* s_msg = ws;
  float* t_msg = s_msg + (size_t)n_src * N_OUT;
  float* es    = t_msg + (size_t)n_tgt * N_OUT;
  float* et    = es + n_src;
  float* e_den = et + n_tgt;               // n_tgt entries
  float* f_den = e_den + n_tgt;            // n_src entries (contiguous after e_den)
  float* e_buf = f_den + n_src;            // n_edges entries

  // Zero outputs + denominators (e_den and f_den are contiguous).
  zero_two<<<1024, 256, 0, stream>>>(msg_src, (long)(n_src + n_tgt) * N_OUT,
                                     e_den, (long)(n_tgt + n_src));

  // Dense projections via fp32 WMMA.
  gemm_wmma_f32<<<(n_src / 16 + 7) / 8, 256, 0, stream>>>(x_source, w_s, s_msg, n_src);
  gemm_wmma_f32<<<(n_tgt / 16 + 7) / 8, 256, 0, stream>>>(x_target, w_t, t_msg, n_tgt);

  // Attention logits: es = s_msg @ att[:64], et = t_msg @ att[64:].
  rowdot64<<<(n_src + 255) / 256, 256, 0, stream>>>(s_msg, att, es, n_src);
  rowdot64<<<(n_tgt + 255) / 256, 256, 0, stream>>>(t_msg, att + N_OUT, et, n_tgt);

  // Edge phase.
  edge_score<<<(n_edges + 255) / 256, 256, 0, stream>>>(rows, cols, es, et,
                                                        e_buf, e_den, f_den, n_edges);
  edge_aggregate<<<(n_edges + 7) / 8, 256, 0, stream>>>(rows, cols, nbhd, e_buf,
                                                        e_den, f_den, s_msg, t_msg,
                                                        msg_src, msg_tgt, n_edges);
}